// LossContrastive_23476291240770
// MI455X (gfx1250) — compile-verified
//
#include <hip/hip_runtime.h>

// LossContrastive, restructured:
//   sim-pair sum   = sum_b (||t_b||^2 - 16)/2,  t_b = sum of 16 normalized views
//   upper-tri sum  = (||s||^2 - 4096)/2,        s   = sum_b t_b
// ||t_b||^2 is computed from the per-sample 16x16 Gram matrix of raw views via
// V_WMMA_F32_16X16X4_F32 (K-accumulated over E=512), scaled by 1/(n_v n_w).
// Total traffic ~9.4 MB -> HBM roofline ~0.4us; no 4096x4096 sim matrix.

typedef __attribute__((ext_vector_type(2))) float v2f;
typedef __attribute__((ext_vector_type(8))) float v8f;

#define BATCH_  256
#define DPTS_   15
#define NVIEW_  16      // text + 15 image views
#define EMBED_  512
#define ROWPAD_ 516     // 516 % 64 == 4 -> conflict-free 16-row column access

__global__ __launch_bounds__(32)
void lc_gram_kernel(const float* __restrict__ text,
                    const float* __restrict__ img,
                    float* __restrict__ t_buf,     // [BATCH][EMBED]
                    float* __restrict__ simraw)    // [BATCH]
{
    __shared__ __align__(16) float rows[NVIEW_][ROWPAD_];  // 33 KB of 320 KB/WGP
    __shared__ float gram[NVIEW_][NVIEW_ + 1];
    __shared__ float rnorm[NVIEW_];

    const int b    = blockIdx.x;
    const int lane = threadIdx.x;   // wave32: one wave per block

    // ---- Stage this sample's 16 rows into LDS (coalesced b128 loads) ----
    for (int v = 0; v < NVIEW_; ++v) {
        const float* src = (v == 0)
            ? (text + (size_t)b * EMBED_)
            : (img  + ((size_t)b * DPTS_ + (v - 1)) * EMBED_);
        for (int j = lane; j < EMBED_ / 4; j += 32) {
            float4 d = reinterpret_cast<const float4*>(src)[j];
            *reinterpret_cast<float4*>(&rows[v][j * 4]) = d;
        }
    }
    __syncthreads();

    // ---- 16x16 Gram via V_WMMA_F32_16X16X4_F32, K looped over 512 ----
    // A (16x4 f32) lane layout: lanes 0-15 = rows M=0..15 holding K={k0,k0+1};
    // lanes 16-31 hold K={k0+2,k0+3}. For a symmetric Gram, B = A^T has the
    // identical register image, so one b64 LDS load feeds both operands.
    const int vrow = lane & 15;
    const float* lp = &rows[vrow][(lane >> 4) * 2];   // 8B-aligned
    v8f c = {};
    #pragma unroll 8
    for (int k0 = 0; k0 < EMBED_; k0 += 4) {
        v2f a = *reinterpret_cast<const v2f*>(lp + k0);
        c = __builtin_amdgcn_wmma_f32_16x16x4_f32(
                false, a, false, a, (short)0, c, false, false);
    }

    // ---- Spill Gram: VGPR r = (M=r, N=lane) lanes0-15, (M=r+8, N=lane-16) hi ----
    const int col   = lane & 15;
    const int rbase = (lane < 16) ? 0 : 8;
    #pragma unroll
    for (int r = 0; r < 8; ++r) gram[rbase + r][col] = c[r];
    __syncthreads();

    if (lane < NVIEW_) rnorm[lane] = 1.0f / sqrtf(gram[lane][lane]);
    __syncthreads();

    // ---- ||t_b||^2 = sum_{v,w} G[v][w] * rn[v] * rn[w] ----
    float part = 0.0f;
    #pragma unroll
    for (int r = 0; r < 8; ++r)
        part += gram[rbase + r][col] * rnorm[rbase + r] * rnorm[col];
    #pragma unroll
    for (int m = 16; m > 0; m >>= 1) part += __shfl_xor(part, m, 32);
    if (lane == 0) simraw[b] = 0.5f * (part - (float)NVIEW_);

    // ---- t_b[e] = sum_v rows[v][e] * rn[v]  (conflict-free column reads) ----
    for (int j = 0; j < EMBED_ / 32; ++j) {
        const int e = lane + 32 * j;
        float t = 0.0f;
        #pragma unroll
        for (int v = 0; v < NVIEW_; ++v) t += rows[v][e] * rnorm[v];
        t_buf[(size_t)b * EMBED_ + e] = t;
    }
}

__global__ __launch_bounds__(512)
void lc_finish_kernel(const float* __restrict__ t_buf,
                      const float* __restrict__ simraw,
                      float* __restrict__ out)
{
    const int tid = threadIdx.x;   // 0..511 == embed index
    float se = 0.0f;
    for (int b = 0; b < BATCH_; ++b) se += t_buf[(size_t)b * EMBED_ + tid];

    float ssq = se * se;
    float sim = (tid < BATCH_) ? simraw[tid] : 0.0f;

    #pragma unroll
    for (int m = 16; m > 0; m >>= 1) {
        ssq += __shfl_xor(ssq, m, 32);
        sim += __shfl_xor(sim, m, 32);
    }
    __shared__ float red[2][16];
    const int wid = tid >> 5;
    if ((tid & 31) == 0) { red[0][wid] = ssq; red[1][wid] = sim; }
    __syncthreads();

    if (tid == 0) {
        float Q = 0.0f, S = 0.0f;
        #pragma unroll
        for (int w = 0; w < 16; ++w) { Q += red[0][w]; S += red[1][w]; }
        const float n      = (float)(BATCH_ * NVIEW_);             // 4096
        const float upper  = 0.5f * (Q - n);
        const float dissim = upper - S;
        const float sim_norm    = S / 30720.0f;    // B * D(D+1)/2
        const float dissim_norm = dissim / 7833600.0f; // D(D+1)B(B-1)/2
        const float loss = 0.5f * dissim_norm - 0.5f * sim_norm; // w_sim = 0.5
        out[0] = loss;
        out[1] = sim_norm;
        out[2] = dissim_norm;
    }
}

extern "C" void kernel_launch(void* const* d_in, const int* in_sizes, int n_in,
                              void* d_out, int out_size, void* d_ws, size_t ws_size,
                              hipStream_t stream) {
    const float* text = (const float*)d_in[0];  // [256,512]
    const float* img  = (const float*)d_in[1];  // [256,15,512]
    float* t_buf  = (float*)d_ws;                              // 256*512 f32
    float* simraw = (float*)d_ws + (size_t)BATCH_ * EMBED_;    // +256 f32 (~513 KB total)

    lc_gram_kernel<<<BATCH_, 32, 0, stream>>>(text, img, t_buf, simraw);
    lc_finish_kernel<<<1, 512, 0, stream>>>(t_buf, simraw, (float*)d_out);
}